// CGINet_36739150250375
// MI455X (gfx1250) — compile-verified
//
#include <hip/hip_runtime.h>
#include <hip/hip_bf16.h>

#define NN 50000
#define E_ADJ 800000
#define E_DEC 200000
#define HH 64

typedef float v2f __attribute__((ext_vector_type(2)));
typedef float v8f __attribute__((ext_vector_type(8)));

// ---------------------------------------------------------------------------
// Dense GEMM: xw[k] = X_k @ W_k   (X_k = x0 or x1, [N,Fin] x [Fin,64])
// grid = (N/16, 4 relations), block = 128 (4 waves; wave w -> col tile w)
// f32 WMMA 16x16x4, K-loop over Fin.
// ---------------------------------------------------------------------------
__global__ __launch_bounds__(128) void gemm_stage(
    const float* __restrict__ x0, const float* __restrict__ x1,
    const float* __restrict__ W, float* __restrict__ xw, int Fin) {
  const int k    = blockIdx.y;           // relation 0..3
  const int tm   = blockIdx.x;           // 16-row tile
  const int wave = threadIdx.x >> 5;     // col tile 0..3 (H=64)
  const int lane = threadIdx.x & 31;
  const int half = lane >> 4;            // 0: K,K+1 | 1: K+2,K+3
  const int l    = lane & 15;

  const float* X  = (k & 1) ? x1 : x0;
  const float* Wk = W + (size_t)k * Fin * HH;

  const int row = tm * 16 + l;           // A-matrix row for this lane
  const int col = wave * 16 + l;         // B/D column for this lane

  const float* ar = X + (size_t)row * Fin + half * 2;
  v8f c = {};
  for (int kk = 0; kk < Fin; kk += 4) {
    v2f a, b;
    a.x = ar[kk];
    a.y = ar[kk + 1];
    const float* wp = Wk + (size_t)(kk + half * 2) * HH + col;
    b.x = wp[0];
    b.y = wp[HH];
    c = __builtin_amdgcn_wmma_f32_16x16x4_f32(
        /*neg_a=*/false, a, /*neg_b=*/false, b,
        /*c_mod=*/(short)0, c, /*reuse_a=*/false, /*reuse_b=*/false);
  }

  float* out = xw + ((size_t)k * NN + (size_t)tm * 16) * HH + wave * 16;
#pragma unroll
  for (int v = 0; v < 8; ++v)
    out[(size_t)(v + half * 8) * HH + l] = c[v];
}

// ---------------------------------------------------------------------------
// Zero fill
// ---------------------------------------------------------------------------
__global__ void zero_f32(float* __restrict__ p, size_t n) {
  size_t i = (size_t)blockIdx.x * blockDim.x + threadIdx.x;
  if (i < n) p[i] = 0.0f;
}

// ---------------------------------------------------------------------------
// SpMM scatter: y[k][rows[e]][:] += vals[e] * xw[k][cols[e]][:]
// 16 threads per edge (float4 each); L2-resident planes; HW float atomics.
// grid = (E_ADJ/16, 4), block = 256
// ---------------------------------------------------------------------------
__global__ __launch_bounds__(256) void spmm_scatter(
    const int* __restrict__ rows, const int* __restrict__ cols,
    const float* __restrict__ vals, const float* __restrict__ xw,
    float* __restrict__ y) {
  const int k = blockIdx.y;
  const int t = threadIdx.x;
  const int e = blockIdx.x * 16 + (t >> 4);
  const int l = t & 15;

  const size_t eo = (size_t)k * E_ADJ + e;
  const int   r = rows[eo];
  const int   cidx = cols[eo];
  const float v = vals[eo];

  const float4 xv =
      ((const float4*)(xw + ((size_t)k * NN + cidx) * HH))[l];
  float* dst = y + ((size_t)k * NN + r) * HH + l * 4;
  unsafeAtomicAdd(dst + 0, xv.x * v);
  unsafeAtomicAdd(dst + 1, xv.y * v);
  unsafeAtomicAdd(dst + 2, xv.z * v);
  unsafeAtomicAdd(dst + 3, xv.w * v);
}

// ---------------------------------------------------------------------------
// h0 = relu(y0)+relu(y1); h1 = relu(y2)+relu(y3)
// ---------------------------------------------------------------------------
__global__ void relu_combine(const float* __restrict__ y,
                             float* __restrict__ h0, float* __restrict__ h1) {
  const size_t S = (size_t)NN * HH;
  size_t i = (size_t)blockIdx.x * blockDim.x + threadIdx.x;
  if (i >= S) return;
  float a = y[i], b = y[S + i], c = y[2 * S + i], d = y[3 * S + i];
  h0[i] = fmaxf(a, 0.0f) + fmaxf(b, 0.0f);
  h1[i] = fmaxf(c, 0.0f) + fmaxf(d, 0.0f);
}

// ---------------------------------------------------------------------------
// M[h][g] = d[h] * R[h][g] * d[g],  d = D_local[rt_k]
// ---------------------------------------------------------------------------
__global__ void build_M(const float* __restrict__ R, const float* __restrict__ D,
                        const int* __restrict__ rt_k, float* __restrict__ M) {
  int i = blockIdx.x * blockDim.x + threadIdx.x;   // 0..4095
  const float* d = D + rt_k[0] * HH;
  int h = i >> 6, g = i & 63;
  M[i] = d[h] * R[i] * d[g];
}

// ---------------------------------------------------------------------------
// Decoder: per 16-edge group compute T = ER(16x64) @ M(64x64) via WMMA,
// then preds[e] = sum_h T[e][h] * EC[e][h].
// grid = E_DEC/16, block = 128 (wave w -> col tile w)
// ---------------------------------------------------------------------------
__global__ __launch_bounds__(128) void decode_edges(
    const float* __restrict__ h0, const float* __restrict__ h1,
    const int* __restrict__ edges, const float* __restrict__ M,
    float* __restrict__ preds) {
  __shared__ float part[4][16];
  const int wave = threadIdx.x >> 5;
  const int lane = threadIdx.x & 31;
  const int half = lane >> 4;
  const int l    = lane & 15;
  const int eb   = blockIdx.x * 16;

  const int arow = edges[(eb + l) * 2 + 0];   // ER row for A layout
  const int col  = wave * 16 + l;

  const float* ar = h0 + (size_t)arow * HH + half * 2;
  v8f c = {};
  for (int kk = 0; kk < HH; kk += 4) {
    v2f a, b;
    a.x = ar[kk];
    a.y = ar[kk + 1];
    const float* mp = M + (kk + half * 2) * HH + col;
    b.x = mp[0];
    b.y = mp[HH];
    c = __builtin_amdgcn_wmma_f32_16x16x4_f32(
        false, a, false, b, (short)0, c, false, false);
  }

  // per-edge dot with EC over this wave's 16 columns
  float p[8];
#pragma unroll
  for (int v = 0; v < 8; ++v) {
    const int edge = v + half * 8;
    const int crow = edges[(eb + edge) * 2 + 1];
    p[v] = c[v] * h1[(size_t)crow * HH + col];
  }
#pragma unroll
  for (int off = 1; off < 16; off <<= 1) {
#pragma unroll
    for (int v = 0; v < 8; ++v) p[v] += __shfl_xor(p[v], off, 16);
  }
  if (l == 0) {
#pragma unroll
    for (int v = 0; v < 8; ++v) part[wave][v + half * 8] = p[v];
  }
  __syncthreads();
  if (threadIdx.x < 16) {
    int t = threadIdx.x;
    preds[eb + t] = part[0][t] + part[1][t] + part[2][t] + part[3][t];
  }
}

// ---------------------------------------------------------------------------
extern "C" void kernel_launch(void* const* d_in, const int* in_sizes, int n_in,
                              void* d_out, int out_size, void* d_ws, size_t ws_size,
                              hipStream_t stream) {
  const int*   adj_rows = (const int*)d_in[0];   // [8, E_ADJ]
  const int*   adj_cols = (const int*)d_in[1];
  const float* adj_vals = (const float*)d_in[2];
  const float* feat0    = (const float*)d_in[3]; // [N,128]
  const float* feat1    = (const float*)d_in[4];
  const float* W1       = (const float*)d_in[5]; // [4,128,64]
  const float* W2       = (const float*)d_in[6]; // [4,64,64]
  const float* W3       = (const float*)d_in[7];
  const float* W4       = (const float*)d_in[8];
  const float* Rg       = (const float*)d_in[9];  // [64,64]
  const float* Dl       = (const float*)d_in[10]; // [4,64]
  const int*   edges    = (const int*)d_in[11];   // [E_DEC,2]
  const int*   rt_k     = (const int*)d_in[12];   // scalar

  const size_t S = (size_t)NN * HH;               // one relation plane
  float* xw = (float*)d_ws;                       // 4*S
  float* y  = xw + 4 * S;                         // 4*S
  float* h0 = y + 4 * S;                          // S
  float* h1 = h0 + S;                             // S
  float* M  = h1 + S;                             // 64*64

  struct Stage { const float* W; int Fin; int base; };
  const Stage st[4] = { {W1, 128, 0}, {W2, 64, 0}, {W3, 64, 4}, {W4, 64, 4} };

  const float* x0 = feat0;
  const float* x1 = feat1;
  for (int s = 0; s < 4; ++s) {
    gemm_stage<<<dim3(NN / 16, 4), 128, 0, stream>>>(x0, x1, st[s].W, xw, st[s].Fin);

    const size_t ycnt = 4 * S;
    zero_f32<<<(unsigned)((ycnt + 255) / 256), 256, 0, stream>>>(y, ycnt);

    spmm_scatter<<<dim3(E_ADJ / 16, 4), 256, 0, stream>>>(
        adj_rows + (size_t)st[s].base * E_ADJ,
        adj_cols + (size_t)st[s].base * E_ADJ,
        adj_vals + (size_t)st[s].base * E_ADJ, xw, y);

    relu_combine<<<(unsigned)((S + 255) / 256), 256, 0, stream>>>(y, h0, h1);
    x0 = h0;
    x1 = h1;
  }

  build_M<<<16, 256, 0, stream>>>(Rg, Dl, rt_k, M);
  decode_edges<<<E_DEC / 16, 128, 0, stream>>>(h0, h1, edges, M, (float*)d_out);
}